// Conv_66803921322223
// MI455X (gfx1250) — compile-verified
//
#include <hip/hip_runtime.h>

#define CIN 32
#define COUT 32
#define KOFF 8

typedef float v2f __attribute__((ext_vector_type(2)));
typedef float v8f __attribute__((ext_vector_type(8)));

__device__ __forceinline__ v8f wmma_f32_16x16x4(v2f a, v2f b, v8f c) {
  // 8 args: (neg_a, A, neg_b, B, c_mod, C, reuse_a, reuse_b)
  return __builtin_amdgcn_wmma_f32_16x16x4_f32(false, a, false, b, (short)0, c,
                                               false, false);
}

// ---------------- pass 0: zero output + counters ----------------
__global__ void spconv_zero(float* __restrict__ out, long n, int* __restrict__ cnt) {
  long i = (long)blockIdx.x * blockDim.x + threadIdx.x;
  long s = (long)gridDim.x * blockDim.x;
  for (long j = i; j < n; j += s) out[j] = 0.0f;
  if (i < 2 * KOFF) cnt[i] = 0;
}

// ---------------- pass 1: histogram over kernel offsets ----------------
__global__ void spconv_hist(const int* __restrict__ offv, int* __restrict__ cnt, int n) {
  int i = blockIdx.x * blockDim.x + threadIdx.x;
  if (i < n) atomicAdd(&cnt[offv[i]], 1);
}

// ---------------- pass 2: tiny exclusive scan (8 bins) ----------------
__global__ void spconv_scan(const int* __restrict__ cnt, int* __restrict__ cur) {
  if (threadIdx.x == 0 && blockIdx.x == 0) {
    int run = 0;
    for (int k = 0; k < KOFF; ++k) {
      cur[k] = run;
      run += cnt[k];
    }
  }
}

// ---------------- pass 3: scatter permutation (bin by offset) ----------------
__global__ void spconv_scatter(const int* __restrict__ offv, int* __restrict__ cur,
                               int* __restrict__ perm, int n) {
  int i = blockIdx.x * blockDim.x + threadIdx.x;
  if (i < n) {
    int pos = atomicAdd(&cur[offv[i]], 1);
    perm[pos] = i;
  }
}

// ---------------- pass 4: WMMA GEMM + scatter-add ----------------
// One wave32 per 16-point tile of the offset-sorted order.
// A (16x32 fp32) lives in 8 chunks of the ISA 16x4 layout:
//   VGPR0: lanes0-15 K=4c, lanes16-31 K=4c+2 ; VGPR1: K=4c+1 / 4c+3
// B (32x32 fp32) from LDS, mirrored layout. D 16x16: VGPR r = rows r / r+8.
__global__ void __launch_bounds__(256)
spconv_wmma(const float* __restrict__ x, const float* __restrict__ w,
            const int* __restrict__ offv, const int* __restrict__ outv,
            const int* __restrict__ perm, float* __restrict__ out,
            int n, int nTiles) {
  __shared__ __align__(16) float lw[KOFF * CIN * COUT];  // 32 KB
  for (int j = threadIdx.x; j < (KOFF * CIN * COUT) / 4; j += blockDim.x)
    ((float4*)lw)[j] = ((const float4*)w)[j];
  __syncthreads();

  const int lane = threadIdx.x & 31;
  const int wave = threadIdx.x >> 5;
  const int half = lane >> 4;  // which K-pair / which D row-half
  const int l16  = lane & 15;
  const int wavesInGrid = gridDim.x * (blockDim.x >> 5);

  for (int tile = blockIdx.x * (blockDim.x >> 5) + wave; tile < nTiles;
       tile += wavesInGrid) {
    const int tileBase = tile * 16;
    const int numValid = min(16, n - tileBase);
    const bool rowValid = l16 < numValid;

    // lane L (both halves) carries metadata of sorted row (L & 15)
    const int pidx = rowValid ? perm[tileBase + l16] : 0;
    const int roff = rowValid ? offv[pidx] : -1;
    const int oidx = rowValid ? outv[pidx] : 0;

    // gather A fragments: one float2 per chunk per lane
    const float* xrow = x + (size_t)pidx * CIN;
    v2f a[8];
#pragma unroll
    for (int c = 0; c < 8; ++c) {
      if (rowValid) {
        const float2 v = *(const float2*)(xrow + 4 * c + 2 * half);
        a[c][0] = v.x;
        a[c][1] = v.y;
      } else {
        a[c][0] = 0.0f;
        a[c][1] = 0.0f;
      }
    }

    // rows are sorted by offset -> a tile spans at most a couple of bins
    const int kmin = __shfl(roff, 0, 32);
    const int kmax = __shfl(roff, numValid - 1, 32);

    v8f acc0 = {};  // output columns 0..15
    v8f acc1 = {};  // output columns 16..31
    for (int k = kmin; k <= kmax; ++k) {  // wave-uniform loop, EXEC stays full
      const float* wk = lw + k * CIN * COUT;
#pragma unroll
      for (int c = 0; c < 8; ++c) {
        const v2f am = (roff == k) ? a[c] : (v2f){0.0f, 0.0f};
        const int rbase = (4 * c + 2 * half) * COUT + l16;
        v2f b0, b1;
        b0[0] = wk[rbase];
        b0[1] = wk[rbase + COUT];
        b1[0] = wk[rbase + 16];
        b1[1] = wk[rbase + 16 + COUT];
        acc0 = wmma_f32_16x16x4(am, b0, acc0);
        acc1 = wmma_f32_16x16x4(am, b1, acc1);
      }
    }

    // scatter-add D: VGPR r holds row r (lanes 0-15) / row r+8 (lanes 16-31)
#pragma unroll
    for (int r = 0; r < 8; ++r) {
      const int m = r + 8 * half;
      const int oi = __shfl(oidx, m, 32);
      const int mv = __shfl(rowValid ? 1 : 0, m, 32);
      if (mv) {
        float* dst = out + (size_t)oi * COUT;
        atomicAdd(dst + l16, acc0[r]);
        atomicAdd(dst + 16 + l16, acc1[r]);
      }
    }
  }
}

extern "C" void kernel_launch(void* const* d_in, const int* in_sizes, int n_in,
                              void* d_out, int out_size, void* d_ws, size_t ws_size,
                              hipStream_t stream) {
  const float* x = (const float*)d_in[0];
  const float* w = (const float*)d_in[1];  // [8,32,32] row-major
  const int* offv = (const int*)d_in[2];
  const int* outv = (const int*)d_in[3];
  float* out = (float*)d_out;

  const int n = in_sizes[0] / CIN;  // number of points
  const int nTiles = (n + 15) / 16;

  int* cnt  = (int*)d_ws;      // [8]
  int* cur  = cnt + KOFF;      // [8]
  int* perm = cnt + 2 * KOFF;  // [n]

  spconv_zero<<<4096, 256, 0, stream>>>(out, (long)out_size, cnt);
  spconv_hist<<<(n + 255) / 256, 256, 0, stream>>>(offv, cnt, n);
  spconv_scan<<<1, 32, 0, stream>>>(cnt, cur);
  spconv_scatter<<<(n + 255) / 256, 256, 0, stream>>>(offv, cur, perm, n);
  spconv_wmma<<<1024, 256, 0, stream>>>(x, w, offv, outv, perm, out, n, nTiles);
}